// RandomMaskSubgraphs_1236950581459
// MI455X (gfx1250) — compile-verified
//
#include <hip/hip_runtime.h>
#include <hip/hip_bf16.h>
#include <stdint.h>

// ---------------------------------------------------------------------------
// RandomMaskSubgraphs on MI455X (gfx1250).
// Memory-bound streaming workload: edge index arrays are DMA'd into LDS with
// the CDNA5 Tensor Data Mover (TENSOR_LOAD_TO_LDS, tracked by TENSORcnt),
// bit-packed node/edge state lives in L2 (25KB/400KB bitmaps << 192MB L2).
// No dense GEMM exists in this reference, so no WMMA; the gfx1250-specific
// path exercised here is the async-tensor (TDM) pipeline.
// ---------------------------------------------------------------------------

#define N_NODES   100000
#define N_EDGES   3200000
#define N_SEEDS   2048
#define TWO_N     200000            // seed ids live in [0, 2*N_NODES)
#define SEED_WORDS 6250             // TWO_N / 32 (exact)
#define EDGE_WORDS 100000           // N_EDGES / 32 (exact)
#define NBUCKETS   4096
#define LIST_CAP   8192
#define TILE       8192             // edges staged per block (2 x 32KB LDS)
#define BLOCK      256              // 8 waves; TILE/32 == BLOCK words/block

typedef unsigned int  v4u __attribute__((ext_vector_type(4)));
typedef int           v8i __attribute__((ext_vector_type(8)));
typedef int           v4i __attribute__((ext_vector_type(4)));

// ---------------------------------------------------------------------------
// TDM: 1-D int32 tile DMA, global -> LDS.  Descriptor layout per CDNA5 ISA
// ch.8 (D# group0/group1).  The tile is expressed as a 2-D tile with
// tile_dim1=1; tensor_dim0 bounds OOB (reads beyond return zero -> ragged
// tail handled for free).  Must be executed by exactly one wave (EXEC is
// ignored by TDM; the guard branch is at wave granularity).
// This toolchain exposes the 6-arg builtin:
//   (uint32x4 g0, int32x8 g1, int32x4 g2, int32x4 g3, int32x8 g4, i32 cpol)
// Groups 2/3/4 are zero (<=2-D tensor, no gather/iterate).
// ---------------------------------------------------------------------------
__device__ __forceinline__ void tdm_load_i32_tile(void* lds_dst,
                                                  const void* gsrc,
                                                  unsigned valid_elems,
                                                  unsigned tile_elems) {
  unsigned long long ga = (unsigned long long)(uintptr_t)gsrc;
  unsigned lds = (unsigned)(uintptr_t)lds_dst;   // low 32 bits = LDS offset

  v4u g0;
  g0[0] = 1u;                                        // count=1 (valid D#)
  g0[1] = lds;                                       // lds_addr
  g0[2] = (unsigned)(ga & 0xFFFFFFFFu);              // global_addr[31:0]
  g0[3] = (unsigned)((ga >> 32) & 0x01FFFFFFu)       // global_addr[56:32]
        | (2u << 30);                                // type = 2 ("image")

  const unsigned td0 = valid_elems;                  // tensor_dim0
  const unsigned td1 = 1u;                           // tensor_dim1
  v8i g1;
  g1[0] = (int)(2u << 16);                           // data_size=2 -> 4 bytes
  g1[1] = (int)((td0 & 0xFFFFu) << 16);              // tensor_dim0[15:0]
  g1[2] = (int)(((td0 >> 16) & 0xFFFFu)              // tensor_dim0[31:16]
        |       ((td1 & 0xFFFFu) << 16));            // tensor_dim1[15:0]
  g1[3] = (int)(((td1 >> 16) & 0xFFFFu)              // tensor_dim1[31:16]
        |       ((tile_elems & 0xFFFFu) << 16));     // tile_dim0
  g1[4] = (int)1u;                                   // tile_dim1=1, tile_dim2=0
  g1[5] = (int)tile_elems;                           // dim0 stride (1 row used)
  g1[6] = 0;
  g1[7] = 0;

  v4i z4; z4[0] = 0; z4[1] = 0; z4[2] = 0; z4[3] = 0;
  v8i z8; z8[0] = 0; z8[1] = 0; z8[2] = 0; z8[3] = 0;
  z8[4] = 0; z8[5] = 0; z8[6] = 0; z8[7] = 0;
  __builtin_amdgcn_tensor_load_to_lds(g0, g1, z4, z4, z8, 0);
}

__device__ __forceinline__ bool testbit(const unsigned* bm, int i) {
  return (bm[i >> 5] >> (i & 31)) & 1u;
}

// Deterministic 32-bit mix hash as the per-(iter,node) uniform score.
__device__ __forceinline__ unsigned hash32(unsigned s, unsigned iter) {
  unsigned x = s * 0x9E3779B9u + (iter + 1u) * 0x85EBCA6Bu;
  x ^= x >> 16; x *= 0x7FEB352Du;
  x ^= x >> 15; x *= 0x846CA68Bu;
  x ^= x >> 16;
  return x;
}

// ---------------------------------------------------------------------------
// Init: zero bitmaps/counters, alive = all-ones (N_EDGES divides 32 exactly).
// ---------------------------------------------------------------------------
__global__ void init_ws(unsigned* seed_mask, unsigned* cand, unsigned* alive,
                        unsigned* masked, unsigned* deg, unsigned* hist,
                        unsigned* ctr) {
  for (int i = blockIdx.x * blockDim.x + threadIdx.x; i < EDGE_WORDS;
       i += gridDim.x * blockDim.x) {
    alive[i]  = 0xFFFFFFFFu;
    masked[i] = 0u;
    if (i < SEED_WORDS) { seed_mask[i] = 0u; cand[i] = 0u; }
    if (i < N_NODES)    deg[i]  = 0u;
    if (i < NBUCKETS)   hist[i] = 0u;
    if (i < 8)          ctr[i]  = 0u;
  }
}

__global__ void clear_u32(unsigned* p, int n) {
  int i = blockIdx.x * blockDim.x + threadIdx.x;
  if (i < n) p[i] = 0u;
}

__global__ void scatter_seeds(const int* __restrict__ seeds,
                              unsigned* seed_mask, int n) {
  int i = blockIdx.x * blockDim.x + threadIdx.x;
  if (i < n) {
    int s = seeds[i];
    atomicOr(&seed_mask[s >> 5], 1u << (s & 31));
  }
}

// ---------------------------------------------------------------------------
// Masking pass: TDM-stage rows/cols tiles, each thread owns one 32-edge word
// of the alive/masked bitmaps (atomic-free word updates).  Candidate set is
// the unique endpoint sums of newly masked edges (bitmap dedupe via atomicOr).
// ---------------------------------------------------------------------------
__global__ __launch_bounds__(BLOCK)
void mask_edges(const int* __restrict__ rows, const int* __restrict__ cols,
                const unsigned* __restrict__ seed_mask,
                unsigned* alive, unsigned* masked, unsigned* cand,
                int make_cand) {
  __shared__ int srows[TILE];
  __shared__ int scols[TILE];

  const long base = (long)blockIdx.x * TILE;
  if (threadIdx.x < 32) {               // wave 0 issues the two DMAs
    unsigned valid = (unsigned)((long)N_EDGES - base < (long)TILE
                                    ? (long)N_EDGES - base : (long)TILE);
    tdm_load_i32_tile(srows, rows + base, valid, TILE);
    tdm_load_i32_tile(scols, cols + base, valid, TILE);
    __builtin_amdgcn_s_wait_tensorcnt(0);
  }
  __syncthreads();

  const int gw = blockIdx.x * BLOCK + threadIdx.x;   // global 32-edge word
  if (gw >= EDGE_WORDS) return;

  unsigned a = alive[gw];
  unsigned hit = 0u;
  const int lbase = threadIdx.x * 32;
  #pragma unroll 4
  for (int b = 0; b < 32; ++b) {
    if (a & (1u << b)) {
      int r = srows[lbase + b];
      int c = scols[lbase + b];
      if (testbit(seed_mask, r) || testbit(seed_mask, c)) hit |= (1u << b);
    }
  }
  if (hit) {
    masked[gw] |= hit;
    alive[gw]   = a & ~hit;
    if (make_cand) {
      unsigned h = hit;
      while (h) {
        int b = __ffs(h) - 1;
        h &= h - 1;
        int s = srows[lbase + b] + scols[lbase + b];   // in [0, TWO_N)
        atomicOr(&cand[s >> 5], 1u << (s & 31));
      }
    }
  }
}

// ---------------------------------------------------------------------------
// Exact-k random selection: popcount -> k, bucket histogram of hash scores,
// scan to threshold bucket, full select below it, exact rank inside it.
// ---------------------------------------------------------------------------
__global__ void count_cand(const unsigned* __restrict__ cand, unsigned* ctr) {
  __shared__ unsigned red[BLOCK];
  unsigned sum = 0u;
  for (int i = blockIdx.x * blockDim.x + threadIdx.x; i < SEED_WORDS;
       i += gridDim.x * blockDim.x)
    sum += __popc(cand[i]);
  red[threadIdx.x] = sum;
  __syncthreads();
  for (int s = BLOCK / 2; s > 0; s >>= 1) {
    if (threadIdx.x < s) red[threadIdx.x] += red[threadIdx.x + s];
    __syncthreads();
  }
  if (threadIdx.x == 0 && red[0]) atomicAdd(&ctr[0], red[0]);
}

__global__ void compute_k(unsigned* ctr, float prob) {
  ctr[1] = (unsigned)floorf((float)ctr[0] * prob);   // k = floor(count * p^(i+1))
}

__global__ void hist_scores(const unsigned* __restrict__ cand,
                            unsigned* hist, int iter) {
  for (int s = blockIdx.x * blockDim.x + threadIdx.x; s < TWO_N;
       s += gridDim.x * blockDim.x)
    if (testbit(cand, s))
      atomicAdd(&hist[hash32((unsigned)s, (unsigned)iter) >> 20], 1u);
}

__global__ void find_threshold(const unsigned* __restrict__ hist,
                               unsigned* ctr) {
  unsigned k = ctr[1];
  int B = -1; unsigned r = 0, cum = 0;
  if (k > 0) {
    for (int b = 0; b < NBUCKETS; ++b) {
      unsigned h = hist[b];
      if (cum + h >= k) { B = b; r = k - cum; break; }
      cum += h;
    }
  }
  ctr[2] = (unsigned)B;
  ctr[3] = r;
  ctr[4] = 0u;          // reset threshold-bucket list count
}

__global__ void select_seeds(const unsigned* __restrict__ cand,
                             unsigned* seed_mask, unsigned* ctr,
                             unsigned long long* list, int iter) {
  int B = (int)ctr[2];
  for (int s = blockIdx.x * blockDim.x + threadIdx.x; s < TWO_N;
       s += gridDim.x * blockDim.x) {
    if (!testbit(cand, s)) continue;
    unsigned h = hash32((unsigned)s, (unsigned)iter);
    int b = (int)(h >> 20);
    if (b < B) {
      atomicOr(&seed_mask[s >> 5], 1u << (s & 31));
    } else if (b == B) {
      unsigned t = atomicAdd(&ctr[4], 1u);
      if (t < LIST_CAP)
        list[t] = ((unsigned long long)h << 32) | (unsigned)s;
    }
  }
}

__global__ void finalize_select(unsigned* seed_mask, const unsigned* ctr,
                                const unsigned long long* __restrict__ list) {
  unsigned m = ctr[4]; if (m > LIST_CAP) m = LIST_CAP;
  unsigned r = ctr[3];
  // exact rank by (score, id): order-independent -> replay-deterministic
  for (unsigned j = threadIdx.x; j < m; j += blockDim.x) {
    unsigned long long key = list[j];
    unsigned rank = 0;
    for (unsigned i = 0; i < m; ++i) rank += (list[i] < key) ? 1u : 0u;
    if (rank < r) {
      int s = (int)(key & 0xFFFFFFFFull);
      atomicOr(&seed_mask[s >> 5], 1u << (s & 31));
    }
  }
}

// ---------------------------------------------------------------------------
// Degree of surviving edges (rows staged via TDM), then D^{-1/2}.
// ---------------------------------------------------------------------------
__global__ __launch_bounds__(BLOCK)
void degree_kernel(const int* __restrict__ rows,
                   const unsigned* __restrict__ alive, unsigned* deg) {
  __shared__ int srows[TILE];
  const long base = (long)blockIdx.x * TILE;
  if (threadIdx.x < 32) {
    unsigned valid = (unsigned)((long)N_EDGES - base < (long)TILE
                                    ? (long)N_EDGES - base : (long)TILE);
    tdm_load_i32_tile(srows, rows + base, valid, TILE);
    __builtin_amdgcn_s_wait_tensorcnt(0);
  }
  __syncthreads();

  const int gw = blockIdx.x * BLOCK + threadIdx.x;
  if (gw >= EDGE_WORDS) return;
  unsigned a = alive[gw];
  const int lbase = threadIdx.x * 32;
  while (a) {
    int b = __ffs(a) - 1;
    a &= a - 1;
    atomicAdd(&deg[srows[lbase + b]], 1u);
  }
}

__global__ void dinv_kernel(const unsigned* __restrict__ deg, float* dinv) {
  int n = blockIdx.x * blockDim.x + threadIdx.x;
  if (n < N_NODES) dinv[n] = rsqrtf((float)deg[n] + 1e-12f);
}

// ---------------------------------------------------------------------------
// Output: enc_vals (float32) then masked flags (int32), concatenated in d_out.
// ---------------------------------------------------------------------------
__global__ __launch_bounds__(BLOCK)
void output_kernel(const int* __restrict__ rows, const int* __restrict__ cols,
                   const unsigned* __restrict__ alive,
                   const unsigned* __restrict__ masked,
                   const float* __restrict__ dinv,
                   float* __restrict__ enc_out, int* __restrict__ mask_out) {
  __shared__ int srows[TILE];
  __shared__ int scols[TILE];
  const long base = (long)blockIdx.x * TILE;
  if (threadIdx.x < 32) {
    unsigned valid = (unsigned)((long)N_EDGES - base < (long)TILE
                                    ? (long)N_EDGES - base : (long)TILE);
    tdm_load_i32_tile(srows, rows + base, valid, TILE);
    tdm_load_i32_tile(scols, cols + base, valid, TILE);
    __builtin_amdgcn_s_wait_tensorcnt(0);
  }
  __syncthreads();

  const int gw = blockIdx.x * BLOCK + threadIdx.x;
  if (gw >= EDGE_WORDS) return;
  unsigned a = alive[gw];
  unsigned mm = masked[gw];
  const int lbase = threadIdx.x * 32;
  const long ebase = (long)gw * 32;
  #pragma unroll 4
  for (int b = 0; b < 32; ++b) {
    float v = 0.f;
    if ((a >> b) & 1u)
      v = dinv[srows[lbase + b]] * dinv[scols[lbase + b]];
    enc_out[ebase + b]  = v;
    mask_out[ebase + b] = (int)((mm >> b) & 1u);
  }
}

// ---------------------------------------------------------------------------
// Host-side orchestration (graph-capture-safe: only kernel launches).
// ---------------------------------------------------------------------------
extern "C" void kernel_launch(void* const* d_in, const int* in_sizes, int n_in,
                              void* d_out, int out_size, void* d_ws,
                              size_t ws_size, hipStream_t stream) {
  const int* adj_rows = (const int*)d_in[0];
  const int* adj_cols = (const int*)d_in[1];
  // d_in[2] (adj_vals) is accepted but unused, matching the reference.
  const int* seeds    = (const int*)d_in[3];

  float* enc_out = (float*)d_out;
  int*   msk_out = (int*)d_out + N_EDGES;

  // Workspace carve-up (~1.73 MB)
  unsigned* ws        = (unsigned*)d_ws;
  unsigned* seed_mask = ws;
  unsigned* cand      = seed_mask + SEED_WORDS;
  unsigned* alive     = cand + SEED_WORDS;
  unsigned* masked    = alive + EDGE_WORDS;
  unsigned* deg       = masked + EDGE_WORDS;
  float*    dinv      = (float*)(deg + N_NODES);
  unsigned* hist      = deg + 2 * N_NODES;
  unsigned* ctr       = hist + NBUCKETS;
  unsigned long long* list = (unsigned long long*)(ctr + 8);

  const int edge_blocks = (EDGE_WORDS + BLOCK - 1) / BLOCK;   // 391
  const int node_blocks = (TWO_N + BLOCK - 1) / BLOCK;        // 782

  init_ws<<<edge_blocks, BLOCK, 0, stream>>>(seed_mask, cand, alive, masked,
                                             deg, hist, ctr);
  scatter_seeds<<<(N_SEEDS + BLOCK - 1) / BLOCK, BLOCK, 0, stream>>>(
      seeds, seed_mask, N_SEEDS);

  const float probs[2] = {0.5f, 0.25f};   // PATH_PROB^(i+1)
  for (int i = 0; i < 3; ++i) {
    mask_edges<<<edge_blocks, BLOCK, 0, stream>>>(
        adj_rows, adj_cols, seed_mask, alive, masked, cand, (i != 2) ? 1 : 0);
    if (i != 2) {
      count_cand<<<32, BLOCK, 0, stream>>>(cand, ctr);
      compute_k<<<1, 1, 0, stream>>>(ctr, probs[i]);
      hist_scores<<<node_blocks, BLOCK, 0, stream>>>(cand, hist, i);
      find_threshold<<<1, 1, 0, stream>>>(hist, ctr);
      clear_u32<<<(SEED_WORDS + BLOCK - 1) / BLOCK, BLOCK, 0, stream>>>(
          seed_mask, SEED_WORDS);
      select_seeds<<<node_blocks, BLOCK, 0, stream>>>(cand, seed_mask, ctr,
                                                      list, i);
      finalize_select<<<1, BLOCK, 0, stream>>>(seed_mask, ctr, list);
      // reset per-iteration scratch for next round
      clear_u32<<<(SEED_WORDS + BLOCK - 1) / BLOCK, BLOCK, 0, stream>>>(
          cand, SEED_WORDS);
      clear_u32<<<(NBUCKETS + BLOCK - 1) / BLOCK, BLOCK, 0, stream>>>(
          hist, NBUCKETS);
      clear_u32<<<1, 8, 0, stream>>>(ctr, 8);
    }
  }

  degree_kernel<<<edge_blocks, BLOCK, 0, stream>>>(adj_rows, alive, deg);
  dinv_kernel<<<(N_NODES + BLOCK - 1) / BLOCK, BLOCK, 0, stream>>>(deg, dinv);
  output_kernel<<<edge_blocks, BLOCK, 0, stream>>>(
      adj_rows, adj_cols, alive, masked, dinv, enc_out, msk_out);
}